// SimpleCarCost_67714454388993
// MI455X (gfx1250) — compile-verified
//
#include <hip/hip_runtime.h>
#include <math.h>

// ---- problem constants (from reference) ----
#define K_TRAJ  4096
#define B_DIM   16
#define T_DIM   64
#define PX_DIM  1024
#define PTS_PER_BLK (B_DIM * T_DIM)   // 1024 points per trajectory k

__device__ __forceinline__ int m2px(float m) {
    // ((m + 32) * 16) truncated toward zero (matches jnp astype(int32)), clipped
    int v = (int)((m + 32.0f) * 16.0f);
    v = v < 0 ? 0 : v;
    return v > (PX_DIM - 1) ? (PX_DIM - 1) : v;
}

__global__ __launch_bounds__(256) void car_cost_kernel(
    const float* __restrict__ state,   // [B, K, T, 12]
    const float* __restrict__ bev,     // [1024, 1024] raw costmap
    const float* __restrict__ path,    // [T, 4]
    float* __restrict__ out)           // [K]
{
    __shared__ float s_state[PTS_PER_BLK * 12];             // 48 KB staged tile
    __shared__ float s_px[T_DIM], s_py[T_DIM];
    __shared__ float s_pc[T_DIM], s_ps[T_DIM], s_pv[T_DIM];
    __shared__ float s_red[8];

    const int k   = blockIdx.x;
    const int tid = threadIdx.x;

    // ---- stage path + precompute heading trig (tiny, broadcast data) ----
    if (tid < T_DIM) {
        float4 p = ((const float4*)path)[tid];
        s_px[tid] = p.x;
        s_py[tid] = p.y;
        s_pc[tid] = __cosf(p.z);
        s_ps[tid] = __sinf(p.z);
        s_pv[tid] = p.w;
    }
    __syncthreads();

    // ---- async-DMA this trajectory's 16x64 state rows (48 B each) into LDS ----
    // Each thread stages 4 points; consecutive tids hit consecutive 48 B rows,
    // so the wave issues fully-coalesced 128 b async copies. INST_OFFSET is
    // added to BOTH the global and LDS addresses, so one base pair covers the
    // 3 x 16 B pieces of a row. th:TH_LOAD_NT keeps this 201 MB use-once
    // stream from evicting the L2-resident 4 MB costmap that the corner
    // gathers depend on.
    #pragma unroll
    for (int i = 0; i < 4; ++i) {
        const int p = tid + i * 256;
        const int b = p >> 6;
        const int t = p & 63;
        const float* gp = state + ((size_t)(b * K_TRAJ + k) * T_DIM + t) * 12;
        unsigned la = (unsigned)(uintptr_t)(s_state + p * 12);  // flat[31:0] == LDS addr
        asm volatile(
            "global_load_async_to_lds_b128 %0, %1, off th:TH_LOAD_NT\n\t"
            "global_load_async_to_lds_b128 %0, %1, off offset:16 th:TH_LOAD_NT\n\t"
            "global_load_async_to_lds_b128 %0, %1, off offset:32 th:TH_LOAD_NT"
            :: "v"(la), "v"(gp) : "memory");
    }
    // Each thread reads back only its own staged bytes -> ASYNCcnt==0 suffices,
    // no workgroup barrier required.
    asm volatile("s_wait_asynccnt 0" ::: "memory");

    // ---- per-point cost, accumulated thread-locally ----
    float acc = 0.0f;
    #pragma unroll
    for (int i = 0; i < 4; ++i) {
        const int p = tid + i * 256;
        const int t = p & 63;
        const float* sp = s_state + p * 12;

        const float x = sp[0],  y     = sp[1];
        const float roll = sp[3], pitch = sp[4], yaw = sp[5];
        const float vx = sp[6], vy    = sp[7];
        const float ay = sp[10], az   = sp[11];

        float syw, cyw;
        __sincosf(yaw, &syw, &cyw);

        // bounding-box corner gathers (L2-resident costmap)
        const float lc = 1.5f  * cyw, ls = 1.5f  * syw;   // CAR_L2
        const float wc = 0.75f * cyw, ws = 0.75f * syw;   // CAR_W2
        const float cxs[4] = { x + lc - ws, x + lc + ws, x - lc - ws, x - lc + ws };
        const float cys[4] = { y + ls + wc, y + ls - wc, y - ls + wc, y - ls - wc };
        float state_cost = 0.0f;
        #pragma unroll
        for (int c = 0; c < 4; ++c) {
            const int px = m2px(cxs[c]);
            const int py = m2px(cys[c]);
            const float g = (255.0f - bev[py * PX_DIM + px]) * (1.0f / 255.0f);
            state_cost = fmaxf(state_cost, g * g);
        }

        // roll/ditch cost (|ay/az| is clipped to [0,1] right after, so the
        // fast rcp-based divide is plenty accurate)
        const float cr = __cosf(roll), cp = __cosf(pitch);
        const float vert = fabsf(az - 9.8f * cr * cp) - 5.0f;
        const float ri   = fabsf(__fdividef(ay, az)) - 1.0f;
        const float rd = (fminf(fmaxf(vert, 0.0f), 10.0f) * 0.1f
                        + fminf(fmaxf(ri,   0.0f), 1.0f)) * 10.0f;

        // tracking costs
        const float sgn  = vx > 0.0f ? 1.0f : (vx < 0.0f ? -1.0f : 0.0f);
        const float V    = sqrtf(vx * vx + vy * vy) * sgn;
        const float beta = atan2f(vy, vx);

        const float xe = x - s_px[t];
        const float ye = y - s_py[t];
        const float ce = cyw - s_pc[t];
        const float se = syw - s_ps[t];
        const float pos_err = xe * xe + ye * ye;
        const float yaw_err = ce * ce + se * se;
        const float ve      = V - s_pv[t];

        const float run = pos_err + yaw_err + 0.5f * ve * ve + 1.5f * beta * beta;
        const float scale = 0.1f + 0.9f * (float)t * (1.0f / 63.0f);  // linspace(0.1, 1, 64)
        const float cons = (pos_err < 1.0f) ? 0.0f : 100.0f * state_cost;

        acc += run * scale + rd + cons;
    }

    // ---- block reduction: wave32 shuffle tree, then 8 partials through LDS ----
    #pragma unroll
    for (int off = 16; off > 0; off >>= 1)
        acc += __shfl_xor(acc, off, 32);

    const int wid  = tid >> 5;
    const int lane = tid & 31;
    if (lane == 0) s_red[wid] = acc;
    __syncthreads();
    if (tid == 0) {
        float s = 0.0f;
        #pragma unroll
        for (int w = 0; w < 8; ++w) s += s_red[w];
        out[k] = s * (1.0f / (float)B_DIM);   // mean over B, sum over T
    }
}

extern "C" void kernel_launch(void* const* d_in, const int* in_sizes, int n_in,
                              void* d_out, int out_size, void* d_ws, size_t ws_size,
                              hipStream_t stream) {
    const float* state = (const float*)d_in[0];
    // d_in[1] = controls: present in the input dict but unused by the reference
    const float* bev   = (const float*)d_in[2];
    const float* path  = (const float*)d_in[3];
    float* out = (float*)d_out;

    car_cost_kernel<<<dim3(K_TRAJ), dim3(256), 0, stream>>>(state, bev, path, out);
}